// GINLayer_12996571038504
// MI455X (gfx1250) — compile-verified
//
#include <hip/hip_runtime.h>
#include <hip/hip_bf16.h>

typedef __attribute__((ext_vector_type(2))) float v2f;
typedef __attribute__((ext_vector_type(8))) float v8f;
typedef __attribute__((ext_vector_type(4))) unsigned int v4u;
typedef __attribute__((ext_vector_type(8))) int v8i;
typedef __attribute__((ext_vector_type(4))) int v4i;

#define BN_EPS 1e-5f

#if __has_builtin(__builtin_amdgcn_tensor_load_to_lds)
#define USE_TDM 1
#else
#define USE_TDM 0
#endif

// ---------------------------------------------------------------------------
// Device-scope FP32 atomic add via CDNA5 inline asm (hardware
// global_atomic_add_f32, no CAS fallback). s_endpgm implies a full wait.
// ---------------------------------------------------------------------------
__device__ __forceinline__ void atomic_fadd_dev(float* addr, float val) {
    asm volatile("global_atomic_add_f32 %0, %1, off scope:SCOPE_DEV"
                 :: "v"(addr), "v"(val)
                 : "memory");
}

#if USE_TDM
// ---------------------------------------------------------------------------
// TDM: DMA a 2D row-major f32 tensor (rowLen x nRows, tile == tensor) from
// global memory into LDS. D# layout per CDNA5 ISA ch.8:
//  group0: [1:0]=count=1, [63:32]=lds_addr, [120:64]=global_addr, [127:126]=2
//  group1: data_size=2(4B), tensor_dim0=tile_dim0=rowLen, tensor_dim1=
//          tile_dim1=nRows, tensor_dim0_stride=rowLen, everything else 0
//  groups 2/3 (+extra group on this toolchain's 6-arg builtin): zero (2D)
// ---------------------------------------------------------------------------
__device__ __forceinline__ void tdm_load_2d_to_lds(const void* g, unsigned ldsOff,
                                                   unsigned rowLen, unsigned nRows) {
    unsigned long long ga = (unsigned long long)(size_t)g;
    v4u g0;
    g0.x = 0x1u;                                            // count=1
    g0.y = ldsOff;                                          // lds_addr (bytes)
    g0.z = (unsigned)(ga & 0xffffffffu);                    // global_addr[31:0]
    g0.w = (unsigned)((ga >> 32) & 0x1ffffffu) | (2u << 30);// addr[56:32] | type=2
    v8i g1;
    g1[0] = 0x00020000;                // data_size=2 (4 bytes), wg_mask=0
    g1[1] = (int)(rowLen << 16);       // tensor_dim0[15:0]
    g1[2] = (int)(nRows << 16);        // tensor_dim1[15:0]
    g1[3] = (int)(rowLen << 16);       // tile_dim0
    g1[4] = (int)nRows;                // tile_dim1
    g1[5] = (int)rowLen;               // tensor_dim0_stride[31:0]
    g1[6] = 0;
    g1[7] = 0;
    v4i z4 = {0, 0, 0, 0};
    v8i z8 = {0, 0, 0, 0, 0, 0, 0, 0};
    __builtin_amdgcn_tensor_load_to_lds(g0, g1, z4, z4, z8, 0);
}
#endif

// ---------------------------------------------------------------------------
// K0: zero a small f32 region (BN stats accumulators)
// ---------------------------------------------------------------------------
__global__ void zero_f32_kernel(float* __restrict__ p, int n) {
    int i = blockIdx.x * blockDim.x + threadIdx.x;
    if (i < n) p[i] = 0.0f;
}

// ---------------------------------------------------------------------------
// K1: combined = (1 + eps) * x   (float4 vectorized)
// ---------------------------------------------------------------------------
__global__ void init_combined_kernel(const float4* __restrict__ x,
                                     const float* __restrict__ epsp,
                                     float4* __restrict__ c, int total4) {
    int i = blockIdx.x * blockDim.x + threadIdx.x;
    if (i >= total4) return;
    float k = 1.0f + *epsp;
    float4 v = x[i];
    v.x *= k; v.y *= k; v.z *= k; v.w *= k;
    c[i] = v;
}

// ---------------------------------------------------------------------------
// K2: edge scatter-add. One thread handles 4 consecutive features of one edge
// (one b128 gather + 4 HW fp32 atomics). 1.6M edges * 16 threads.
// x and combined both fit in the 192MB L2, so RMWs stay on-chip.
// ---------------------------------------------------------------------------
__global__ void edge_scatter_kernel(const float* __restrict__ x,
                                    const int* __restrict__ ei,
                                    float* __restrict__ combined,
                                    int n_edges) {
    unsigned idx = blockIdx.x * blockDim.x + threadIdx.x;
    unsigned total = (unsigned)n_edges * 16u;
    if (idx >= total) return;
    unsigned e = idx >> 4;
    int d = (int)(idx & 15u) * 4;
    int s = ei[e];                      // src row
    int t = ei[(unsigned)n_edges + e];  // dst row
    float4 v = *(const float4*)(x + (size_t)s * 64 + d);
    float* dst = combined + (size_t)t * 64 + d;
    atomic_fadd_dev(dst,     v.x);
    atomic_fadd_dev(dst + 1, v.y);
    atomic_fadd_dev(dst + 2, v.z);
    atomic_fadd_dev(dst + 3, v.w);
}

// ---------------------------------------------------------------------------
// Stage W into LDS: TDM from wave 0 (TENSORcnt wait) or fallback copy loop.
// Must run before any wave-uniform early exit (all waves hit the barrier).
// ---------------------------------------------------------------------------
__device__ __forceinline__ void stage_weights(const float* W, float* ldsW,
                                              int rowLen, int nRows) {
#if USE_TDM
    if (threadIdx.x < 32) {
        tdm_load_2d_to_lds(W, (unsigned)(size_t)ldsW,
                           (unsigned)rowLen, (unsigned)nRows);
        __builtin_amdgcn_s_wait_tensorcnt(0);
    }
#else
    for (int i = threadIdx.x; i < rowLen * nRows; i += blockDim.x) ldsW[i] = W[i];
#endif
    __syncthreads();
}

// ---------------------------------------------------------------------------
// K3: GEMM1  h1[N,128] = combined[N,64] @ W1[64,128] + b1
// One wave per 16-row tile; V_WMMA_F32_16X16X4_F32, 16 K-steps, 8 col tiles.
// A fragments preloaded (32 VGPRs); B fragments served from LDS (ds_load).
// A layout (16x4 f32): lane l<16 -> A[M=l][K=0..1]; lane l+16 -> K=2..3.
// C/D: vgpr r, lanes 0-15 -> M=r, lanes 16-31 -> M=r+8; N = lane%16.
// ---------------------------------------------------------------------------
__global__ void gemm1_kernel(const float* __restrict__ A,   // [N,64]
                             const float* __restrict__ W,   // [64,128]
                             const float* __restrict__ bias,// [128]
                             float* __restrict__ H,         // [N,128]
                             int mtiles) {
    __shared__ float ldsW[64 * 128];
    stage_weights(W, ldsW, 128, 64);    // tensor 128-wide x 64 rows

    const int lane = threadIdx.x & 31;
    const int wave = blockIdx.x * (blockDim.x >> 5) + (threadIdx.x >> 5);
    if (wave >= mtiles) return;         // whole-wave uniform exit
    const int row0 = wave * 16;
    const int half = lane >> 4;         // 0 or 1
    const int l16  = lane & 15;

    // Preload 16 A fragments (K = 4*kk .. 4*kk+3)
    v2f a[16];
    const float* aptr = A + (size_t)(row0 + l16) * 64 + half * 2;
#pragma unroll
    for (int kk = 0; kk < 16; ++kk)
        a[kk] = *(const v2f*)(aptr + kk * 4);

#pragma unroll
    for (int nt = 0; nt < 8; ++nt) {
        const int ncol = nt * 16 + l16;
        float bv = bias[ncol];
        v8f acc;
#pragma unroll
        for (int r = 0; r < 8; ++r) acc[r] = bv;   // C[m][n] = b1[n]

        const float* wp = ldsW + (size_t)(half * 2) * 128 + ncol;
#pragma unroll
        for (int kk = 0; kk < 16; ++kk) {
            v2f b;
            b.x = wp[(size_t)kk * 512];            // row K=4kk+half*2
            b.y = wp[(size_t)kk * 512 + 128];      // row K=4kk+half*2+1
            acc = __builtin_amdgcn_wmma_f32_16x16x4_f32(
                false, a[kk], false, b, (short)0, acc, false, false);
        }
        float* out = H + (size_t)(row0 + half * 8) * 128 + ncol;
#pragma unroll
        for (int r = 0; r < 8; ++r)
            out[(size_t)r * 128] = acc[r];
    }
}

// ---------------------------------------------------------------------------
// K4/K7: per-feature sum & sum-of-squares over rows. F divides blockDim.
// ---------------------------------------------------------------------------
__global__ void reduce_stats_kernel(const float* __restrict__ H, int nrows,
                                    int F, float* __restrict__ stats) {
    int f    = threadIdx.x & (F - 1);
    int rpb  = blockDim.x / F;
    int rloc = threadIdx.x / F;
    float s = 0.0f, ss = 0.0f;
    for (int r = blockIdx.x * rpb + rloc; r < nrows; r += gridDim.x * rpb) {
        float v = H[(size_t)r * F + f];
        s += v; ss += v * v;
    }
    atomic_fadd_dev(&stats[f], s);
    atomic_fadd_dev(&stats[F + f], ss);
}

// ---------------------------------------------------------------------------
// K5: fold stats into BN scale/shift:  y = scale*h + shift
// ---------------------------------------------------------------------------
__global__ void bn_params_kernel(const float* __restrict__ stats,
                                 const float* __restrict__ gamma,
                                 const float* __restrict__ beta,
                                 float* __restrict__ bnp, int F, float invN) {
    int f = blockIdx.x * blockDim.x + threadIdx.x;
    if (f >= F) return;
    float mean = stats[f] * invN;
    float var  = stats[F + f] * invN - mean * mean;
    float sc   = gamma[f] * rsqrtf(var + BN_EPS);
    bnp[f]     = sc;
    bnp[F + f] = beta[f] - mean * sc;
}

// ---------------------------------------------------------------------------
// K6: GEMM2  out[N,64] = relu(bn1(h1))[N,128] @ W2[128,64] + b2
// BN1 + ReLU fused into the A-fragment load. 32 K-steps, 4 col tiles.
// W2 staged to LDS via TDM; B fragments from LDS.
// ---------------------------------------------------------------------------
__global__ void gemm2_kernel(const float* __restrict__ H1,   // [N,128]
                             const float* __restrict__ W,    // [128,64]
                             const float* __restrict__ bias, // [64]
                             const float* __restrict__ bnp1, // [256] scale|shift
                             float* __restrict__ OUT,        // [N,64]
                             int mtiles) {
    __shared__ float ldsW[128 * 64];
    stage_weights(W, ldsW, 64, 128);    // tensor 64-wide x 128 rows

    const int lane = threadIdx.x & 31;
    const int wave = blockIdx.x * (blockDim.x >> 5) + (threadIdx.x >> 5);
    if (wave >= mtiles) return;
    const int row0 = wave * 16;
    const int half = lane >> 4;
    const int l16  = lane & 15;

    // Preload + normalize 32 A fragments (features f0 = 4kk+half*2, f0+1)
    v2f a[32];
    const float* aptr = H1 + (size_t)(row0 + l16) * 128 + half * 2;
#pragma unroll
    for (int kk = 0; kk < 32; ++kk) {
        int f0 = kk * 4 + half * 2;
        v2f h  = *(const v2f*)(aptr + kk * 4);
        v2f sc = *(const v2f*)(bnp1 + f0);
        v2f sh = *(const v2f*)(bnp1 + 128 + f0);
        a[kk].x = fmaxf(0.0f, h.x * sc.x + sh.x);
        a[kk].y = fmaxf(0.0f, h.y * sc.y + sh.y);
    }

#pragma unroll
    for (int nt = 0; nt < 4; ++nt) {
        const int ncol = nt * 16 + l16;
        float bv = bias[ncol];
        v8f acc;
#pragma unroll
        for (int r = 0; r < 8; ++r) acc[r] = bv;

        const float* wp = ldsW + (size_t)(half * 2) * 64 + ncol;
#pragma unroll
        for (int kk = 0; kk < 32; ++kk) {
            v2f b;
            b.x = wp[(size_t)kk * 256];
            b.y = wp[(size_t)kk * 256 + 64];
            acc = __builtin_amdgcn_wmma_f32_16x16x4_f32(
                false, a[kk], false, b, (short)0, acc, false, false);
        }
        float* out = OUT + (size_t)(row0 + half * 8) * 64 + ncol;
#pragma unroll
        for (int r = 0; r < 8; ++r)
            out[(size_t)r * 64] = acc[r];
    }
}

// ---------------------------------------------------------------------------
// K8: in-place BN2 on d_out
// ---------------------------------------------------------------------------
__global__ void bn_apply_kernel(float* __restrict__ H,
                                const float* __restrict__ stats,
                                const float* __restrict__ gamma,
                                const float* __restrict__ beta,
                                int F, float invN, unsigned total) {
    unsigned i = blockIdx.x * blockDim.x + threadIdx.x;
    if (i >= total) return;
    int f = (int)(i & (unsigned)(F - 1));
    float mean = stats[f] * invN;
    float var  = stats[F + f] * invN - mean * mean;
    float v = H[i];
    H[i] = gamma[f] * (v - mean) * rsqrtf(var + BN_EPS) + beta[f];
}

// ---------------------------------------------------------------------------
extern "C" void kernel_launch(void* const* d_in, const int* in_sizes, int n_in,
                              void* d_out, int out_size, void* d_ws, size_t ws_size,
                              hipStream_t stream) {
    const float* x    = (const float*)d_in[0];
    const int*   ei   = (const int*)  d_in[1];
    const float* epsp = (const float*)d_in[2];
    const float* W1   = (const float*)d_in[3];
    const float* b1   = (const float*)d_in[4];
    const float* g1   = (const float*)d_in[5];
    const float* be1  = (const float*)d_in[6];
    const float* W2   = (const float*)d_in[7];
    const float* b2   = (const float*)d_in[8];
    const float* g2   = (const float*)d_in[9];
    const float* be2  = (const float*)d_in[10];
    float* out = (float*)d_out;

    const int N      = in_sizes[0] / 64;    // 100000 (divisible by 16)
    const int nE     = in_sizes[1] / 2;     // 1.6M
    const int mtiles = N / 16;              // 6250
    const float invN = 1.0f / (float)N;

    // Workspace layout (fp32):
    char* ws = (char*)d_ws;
    float* combined = (float*)ws;                               // N*64
    float* h1       = (float*)(ws + (size_t)N * 64 * 4);        // N*128
    float* stats1   = (float*)(ws + (size_t)N * 192 * 4);       // 256
    float* stats2   = stats1 + 256;                             // 128
    float* bnp1     = stats2 + 128;                             // 256

    // K0: zero BN stats accumulators (ws is not re-poisoned between replays)
    zero_f32_kernel<<<1, 1024, 0, stream>>>(stats1, 256 + 128 + 256);

    // K1: combined = (1+eps)*x
    {
        int total4 = N * 16;  // N*64/4
        int blocks = (total4 + 255) / 256;
        init_combined_kernel<<<blocks, 256, 0, stream>>>(
            (const float4*)x, epsp, (float4*)combined, total4);
    }

    // K2: scatter-add over edges (device-scope HW fp32 atomics)
    {
        unsigned total = (unsigned)nE * 16u;
        unsigned blocks = (total + 255u) / 256u;
        edge_scatter_kernel<<<blocks, 256, 0, stream>>>(x, ei, combined, nE);
    }

    // K3: h1 = combined @ W1 + b1   (fp32 WMMA, W1 staged via TDM)
    {
        int blocks = (mtiles + 7) / 8;      // 8 waves / block
        gemm1_kernel<<<blocks, 256, 0, stream>>>(combined, W1, b1, h1, mtiles);
    }

    // K4: BN1 stats
    reduce_stats_kernel<<<1024, 256, 0, stream>>>(h1, N, 128, stats1);

    // K5: fold into scale/shift
    bn_params_kernel<<<1, 128, 0, stream>>>(stats1, g1, be1, bnp1, 128, invN);

    // K6: out = relu(bn1(h1)) @ W2 + b2   (fused BN+ReLU, W2 staged via TDM)
    {
        int blocks = (mtiles + 7) / 8;
        gemm2_kernel<<<blocks, 256, 0, stream>>>(h1, W2, b2, bnp1, out, mtiles);
    }

    // K7: BN2 stats over output
    reduce_stats_kernel<<<1024, 256, 0, stream>>>(out, N, 64, stats2);

    // K8: in-place BN2
    {
        unsigned total = (unsigned)N * 64u;
        unsigned blocks = (total + 255u) / 256u;
        bn_apply_kernel<<<blocks, 256, 0, stream>>>(out, stats2, g2, be2,
                                                    64, invN, total);
    }
}